// FilterDetections_34815004902006
// MI455X (gfx1250) — compile-verified
//
#include <hip/hip_runtime.h>
#include <stdint.h>

#define B_IMG     8
#define A_NUM     100000
#define C_CLS     80
#define PRE_NMSN  1000
#define POST_NMSN 300
#define NMS_TH    0.5f
#define SCORE_TH  0.3f
#define BBOX_CLIP 4.135166556742356f   /* log(1000/16) */
#define NBINS     2048
#define NGRP      (NBINS / 4)
#define CAND_CAP  2048
#define BLK       256

typedef float v2f __attribute__((ext_vector_type(2)));
typedef float v8f __attribute__((ext_vector_type(8)));

// ---------------------------------------------------------------------------
// Kernel 1: decode anchors + deltas -> boxes (B, A, 4) into workspace
// ---------------------------------------------------------------------------
__global__ __launch_bounds__(256)
void fd_boxes_kernel(const float* __restrict__ reg,
                     const float* __restrict__ anchors,
                     float* __restrict__ boxes, int total)
{
    int t = blockIdx.x * blockDim.x + threadIdx.x;
    if (t >= total) return;
    int a = t % A_NUM;
    float4 anc = ((const float4*)anchors)[a];
    float4 d   = ((const float4*)reg)[t];
    float wa  = anc.z - anc.x;
    float ha  = anc.w - anc.y;
    float cxa = anc.x + 0.5f * wa;
    float cya = anc.y + 0.5f * ha;
    float dw  = fminf(d.z, BBOX_CLIP);
    float dh  = fminf(d.w, BBOX_CLIP);
    float cx  = d.x * wa + cxa;
    float cy  = d.y * ha + cya;
    float w   = expf(dw) * wa;
    float h   = expf(dh) * ha;
    float4 o;
    o.x = cx - 0.5f * w; o.y = cy - 0.5f * h;
    o.z = cx + 0.5f * w; o.w = cy + 0.5f * h;
    ((float4*)boxes)[t] = o;
}

// ---------------------------------------------------------------------------
// Kernel 2: per (image, class) block: top-1000 select + greedy NMS + top-300
// ---------------------------------------------------------------------------
__global__ __launch_bounds__(BLK)
void fd_nms_kernel(const float* __restrict__ cls,
                   const float* __restrict__ boxesWS,
                   float* __restrict__ out)
{
    const int c   = blockIdx.x;
    const int b   = blockIdx.y;
    const int tid = threadIdx.x;

    __shared__ unsigned           hist[NBINS];
    __shared__ float              partF[NGRP];       // 4-bin group sums (via WMMA)
    __shared__ unsigned long long keys[CAND_CAP];
    __shared__ float4             sBox[PRE_NMSN];
    __shared__ float              sArea[PRE_NMSN];
    __shared__ unsigned           sKeep[PRE_NMSN];
    __shared__ unsigned short     sRank[PRE_NMSN];
    __shared__ unsigned           sCnt;
    __shared__ int                sCutBin;
    __shared__ int                sKept;

    // ---- Phase A: score histogram over [SCORE_TH, 1.0) ----
    for (int i = tid; i < NBINS; i += BLK) hist[i] = 0u;
    if (tid == 0) sCnt = 0u;
    __syncthreads();

    const float  invw = (float)NBINS / (1.0f - SCORE_TH);
    const float* scol = cls + (size_t)b * A_NUM * C_CLS + c;

    for (int a = tid; a < A_NUM; a += BLK) {
        __builtin_prefetch(scol + (size_t)(a + BLK) * C_CLS, 0, 0); // global_prefetch_b8
        float s = scol[(size_t)a * C_CLS];
        if (s >= SCORE_TH) {
            int bin = (int)((s - SCORE_TH) * invw);
            bin = bin < NBINS - 1 ? bin : NBINS - 1;
            atomicAdd(&hist[bin], 1u);
        }
    }
    __syncthreads();

    // ---- Phase B1: group-of-4 partial sums via matrix pipe ----
    // part[g] = sum(hist[4g..4g+3]) computed as D = A(16x4) x ones(4x16):
    // every row of D is the 4-bin sum (counts <= 1e5 are exact in f32).
    {
        const int wave = tid >> 5;          // 8 waves
        const int lane = tid & 31;          // wave32
        const int M    = lane & 15;
        const int kh   = (lane < 16) ? 0 : 2;   // ISA 32-bit A layout: K pairs
        v2f bones; bones.x = 1.0f; bones.y = 1.0f;
        for (int w = 0; w < 4; ++w) {
            const int m_base = (wave * 4 + w) * 16;   // 16 groups per WMMA
            v2f a;
            a.x = (float)hist[4 * (m_base + M) + kh];
            a.y = (float)hist[4 * (m_base + M) + kh + 1];
            v8f cacc = {};
            v8f d = __builtin_amdgcn_wmma_f32_16x16x4_f32(
                false, a, false, bones, (short)0, cacc, false, false);
            // D layout: VGPR r -> M=r (lanes 0-15) / M=8+r (lanes 16-31)
            if (lane == 0) {
                #pragma unroll
                for (int r = 0; r < 8; ++r) partF[m_base + r] = d[r];
            } else if (lane == 16) {
                #pragma unroll
                for (int r = 0; r < 8; ++r) partF[m_base + 8 + r] = d[r];
            }
        }
    }
    __syncthreads();

    // ---- Phase B2: find cutoff bin (score of the PRE_NMS-th candidate) ----
    if (tid == 0) {
        float accF = 0.0f; int cut = 0; bool found = false;
        for (int g = NGRP - 1; g >= 0 && !found; --g) {
            float naccF = accF + partF[g];
            if (naccF >= (float)PRE_NMSN) {
                unsigned a2 = (unsigned)accF;          // exact (integral < 2^24)
                cut = 4 * g;
                for (int k = 4 * g + 3; k >= 4 * g; --k) {
                    a2 += hist[k];
                    if (a2 >= PRE_NMSN) { cut = k; break; }
                }
                found = true;
            } else {
                accF = naccF;
            }
        }
        sCutBin = found ? cut : 0;   // fewer than PRE_NMS above thresh -> take all
    }
    __syncthreads();

    // ---- Phase C: collect candidates as sortable 64-bit keys ----
    const int cutBin = sCutBin;
    for (int a = tid; a < A_NUM; a += BLK) {
        float s = scol[(size_t)a * C_CLS];
        if (s >= SCORE_TH) {
            int bin = (int)((s - SCORE_TH) * invw);
            bin = bin < NBINS - 1 ? bin : NBINS - 1;
            if (bin >= cutBin) {
                unsigned pos = atomicAdd(&sCnt, 1u);
                if (pos < CAND_CAP) {
                    // high: score bits (positive floats compare as uints)
                    // low : ~idx  -> ties resolve to smallest idx first (top_k rule)
                    keys[pos] = ((unsigned long long)__float_as_uint(s) << 32)
                              | (unsigned)~(unsigned)a;
                }
            }
        }
    }
    __syncthreads();

    int cnt = (int)sCnt; if (cnt > CAND_CAP) cnt = CAND_CAP;
    for (int i = tid; i < CAND_CAP; i += BLK)
        if (i >= cnt) keys[i] = 0ull;   // pad = -inf sentinel (sorts last)
    __syncthreads();

    // ---- Phase D: bitonic sort, descending ----
    for (unsigned k = 2; k <= CAND_CAP; k <<= 1) {
        for (unsigned j = k >> 1; j > 0; j >>= 1) {
            for (unsigned t = tid; t < CAND_CAP; t += BLK) {
                unsigned ixj = t ^ j;
                if (ixj > t) {
                    unsigned long long x = keys[t], y = keys[ixj];
                    bool desc = ((t & k) == 0);
                    if (desc ? (x < y) : (x > y)) { keys[t] = y; keys[ixj] = x; }
                }
            }
            __syncthreads();
        }
    }

    const int topN = cnt < PRE_NMSN ? cnt : PRE_NMSN;

    // ---- Phase E: gather top-N boxes into LDS via CDNA5 async DMA ----
    for (int jj = tid; jj < PRE_NMSN; jj += BLK) {
        if (jj < topN) {
            unsigned aidx = ~(unsigned)(keys[jj] & 0xffffffffull);
            const float* src = boxesWS + ((size_t)b * A_NUM + aidx) * 4;
            unsigned ldsoff = (unsigned)(uintptr_t)(&sBox[jj]);          // LDS byte offset
            unsigned long long gaddr = (unsigned long long)(uintptr_t)src;
            asm volatile("global_load_async_to_lds_b128 %0, %1, off"
                         :: "v"(ldsoff), "v"(gaddr) : "memory");
            sKeep[jj] = 1u;
        } else {
            sKeep[jj] = 0u;
        }
    }
    asm volatile("s_wait_asynccnt 0" ::: "memory");   // this wave's DMAs landed in LDS
    __syncthreads();                                   // visible to whole workgroup

    for (int jj = tid; jj < topN; jj += BLK) {
        float4 bx = sBox[jj];
        sArea[jj] = fmaxf(bx.z - bx.x, 0.0f) * fmaxf(bx.w - bx.y, 0.0f);
    }

    // ---- Phase F: exact greedy NMS (matches reference fori_loop) ----
    for (int i = 0; i < topN; ++i) {
        __syncthreads();                 // finalize keep[i] from earlier rounds
        if (sKeep[i]) {
            float4 bi = sBox[i];
            float  ai = fmaxf(bi.z - bi.x, 0.0f) * fmaxf(bi.w - bi.y, 0.0f);
            for (int jj = i + 1 + tid; jj < topN; jj += BLK) {
                if (!sKeep[jj]) continue;
                float4 bj  = sBox[jj];
                float xx1 = fmaxf(bi.x, bj.x), yy1 = fmaxf(bi.y, bj.y);
                float xx2 = fminf(bi.z, bj.z), yy2 = fminf(bi.w, bj.w);
                float inter = fmaxf(xx2 - xx1, 0.0f) * fmaxf(yy2 - yy1, 0.0f);
                float iou = inter / (ai + sArea[jj] - inter + 1e-9f);
                if (iou > NMS_TH) sKeep[jj] = 0u;
            }
        }
    }
    __syncthreads();

    // ---- Phase G: compact (kept preserve score order == post top-k) ----
    if (tid == 0) {
        int r = 0;
        for (int jj = 0; jj < topN; ++jj) {
            sRank[jj] = (unsigned short)r;
            r += sKeep[jj] ? 1 : 0;
        }
        sKept = r;
    }
    __syncthreads();

    int kept = sKept; if (kept > POST_NMSN) kept = POST_NMSN;
    float* outBase = out + ((size_t)(b * C_CLS + c)) * POST_NMSN * 6;

    for (int jj = tid; jj < topN; jj += BLK) {
        if (sKeep[jj]) {
            int r = sRank[jj];
            if (r < POST_NMSN) {
                float4 bx = sBox[jj];
                float  sc = __uint_as_float((unsigned)(keys[jj] >> 32));
                float* o = outBase + (size_t)r * 6;
                o[0] = bx.x; o[1] = bx.y; o[2] = bx.z; o[3] = bx.w;
                o[4] = sc;   o[5] = (float)c;
            }
        }
    }
    for (int r = kept + tid; r < POST_NMSN; r += BLK) {
        float* o = outBase + (size_t)r * 6;
        o[0] = 0.0f; o[1] = 0.0f; o[2] = 0.0f; o[3] = 0.0f;
        o[4] = 0.0f; o[5] = -1.0f;
    }
}

// ---------------------------------------------------------------------------
extern "C" void kernel_launch(void* const* d_in, const int* in_sizes, int n_in,
                              void* d_out, int out_size, void* d_ws, size_t ws_size,
                              hipStream_t stream)
{
    const float* cls     = (const float*)d_in[0];  // (8, 100000, 80)
    const float* reg     = (const float*)d_in[1];  // (8, 100000, 4)
    const float* anchors = (const float*)d_in[2];  // (100000, 4)
    float*       out     = (float*)d_out;          // (8, 80, 300, 6)
    float*       boxesWS = (float*)d_ws;           // needs 8*100000*4 floats = 12.8 MB

    const int total = B_IMG * A_NUM;
    fd_boxes_kernel<<<(total + 255) / 256, 256, 0, stream>>>(reg, anchors, boxesWS, total);

    dim3 grid(C_CLS, B_IMG);
    fd_nms_kernel<<<grid, BLK, 0, stream>>>(cls, boxesWS, out);
}